// Model_2619930051526
// MI455X (gfx1250) — compile-verified
//
#include <hip/hip_runtime.h>

// Problem constants (from reference)
#define QL   1024
#define KL   1024
#define NB   64      // H*M = 8*8 batches
#define DD   128
#define OUTK 1025    // K_LEN + 1 sink column
#define SM_SCALE 0.08838834764831845f

typedef __attribute__((ext_vector_type(2))) float v2f;
typedef __attribute__((ext_vector_type(8))) float v8f;

// Q layout: [q, h, m, d]  -> Q[(q*64 + b)*128 + d],  b = h*8+m
// K layout: [k, h, m, d]  -> K[(k*64 + b)*128 + d]
// out layout: [h, m, q, k] -> out[(b*1024 + q)*1025 + k]
// mask: [q, k] -> mask[q*1024 + k]
// sinks: [h, m, q, 1] -> sinks[b*1024 + q]

__device__ __forceinline__ void store_tile(const v8f& c, int b, int qb, int kb,
                                           int qi, int ki, int half, int l16,
                                           const float* __restrict__ mask,
                                           float* __restrict__ out)
{
    const int kcol = kb + ki * 16 + l16;
#pragma unroll
    for (int r = 0; r < 8; ++r) {
        const int qrow = qb + qi * 16 + half * 8 + r;
        float v = c[r] * SM_SCALE + mask[(size_t)qrow * KL + kcol];
        __builtin_nontemporal_store(v, out + ((size_t)b * QL + qrow) * OUTK + kcol);
    }
}

__global__ __launch_bounds__(256) void qk_wmma_kernel(
    const float* __restrict__ Q, const float* __restrict__ K,
    const float* __restrict__ mask, float* __restrict__ out)
{
    const int lane = threadIdx.x & 31;
    const int wave = threadIdx.x >> 5;   // 0..7
    const int half = lane >> 4;          // which 16-lane half
    const int l16  = lane & 15;

    const int b  = blockIdx.z;                             // batch (h*8+m)
    const int kb = blockIdx.x * 128 + (wave & 3) * 32;     // key tile base
    const int qb = blockIdx.y * 64  + (wave >> 2) * 32;    // query tile base

    // Per-lane row base pointers; the +2*half matches the 16x4 f32 WMMA
    // fragment layout (lanes 0-15 carry K=0,1 / lanes 16-31 carry K=2,3).
    const float* qp0 = Q + ((size_t)(qb + l16     ) * 64 + b) * DD + 2 * half;
    const float* qp1 = Q + ((size_t)(qb + 16 + l16) * 64 + b) * DD + 2 * half;
    const float* kp0 = K + ((size_t)(kb + l16     ) * 64 + b) * DD + 2 * half;
    const float* kp1 = K + ((size_t)(kb + 16 + l16) * 64 + b) * DD + 2 * half;

    v8f c00 = {}, c01 = {}, c10 = {}, c11 = {};

#pragma unroll 8
    for (int d0 = 0; d0 < DD; d0 += 4) {
        v2f a0 = *(const v2f*)(qp0 + d0);
        v2f a1 = *(const v2f*)(qp1 + d0);
        v2f b0 = *(const v2f*)(kp0 + d0);
        v2f b1 = *(const v2f*)(kp1 + d0);
        c00 = __builtin_amdgcn_wmma_f32_16x16x4_f32(false, a0, false, b0, (short)0, c00, false, false);
        c01 = __builtin_amdgcn_wmma_f32_16x16x4_f32(false, a0, false, b1, (short)0, c01, false, false);
        c10 = __builtin_amdgcn_wmma_f32_16x16x4_f32(false, a1, false, b0, (short)0, c10, false, false);
        c11 = __builtin_amdgcn_wmma_f32_16x16x4_f32(false, a1, false, b1, (short)0, c11, false, false);
    }

    // Fused epilogue: scale, add mask, non-temporal streaming store
    // (output is 268.7 MB > 192 MB L2 -> bypass with NT hint).
    store_tile(c00, b, qb, kb, 0, 0, half, l16, mask, out);
    store_tile(c01, b, qb, kb, 0, 1, half, l16, mask, out);
    store_tile(c10, b, qb, kb, 1, 0, half, l16, mask, out);
    store_tile(c11, b, qb, kb, 1, 1, half, l16, mask, out);
}

// Writes the sink column k = 1024: out[(b*1024+q)*1025 + 1024] = sinks[b*1024+q]
__global__ __launch_bounds__(256) void sink_kernel(
    const float* __restrict__ sinks, float* __restrict__ out)
{
    const int idx = blockIdx.x * blockDim.x + threadIdx.x; // 0 .. 65535
    __builtin_nontemporal_store(sinks[idx], out + (size_t)idx * OUTK + (OUTK - 1));
}

extern "C" void kernel_launch(void* const* d_in, const int* in_sizes, int n_in,
                              void* d_out, int out_size, void* d_ws, size_t ws_size,
                              hipStream_t stream)
{
    const float* Q     = (const float*)d_in[0];
    const float* Kx    = (const float*)d_in[1];
    const float* mask  = (const float*)d_in[2];
    const float* sinks = (const float*)d_in[3];
    float* out = (float*)d_out;

    dim3 grid(KL / 128, QL / 64, NB);   // 8 x 16 x 64 = 8192 blocks
    qk_wmma_kernel<<<grid, 256, 0, stream>>>(Q, Kx, mask, out);

    sink_kernel<<<(NB * QL) / 256, 256, 0, stream>>>(sinks, out);
}